// PillarFeatureNetScatter_71313636983307
// MI455X (gfx1250) — compile-verified
//
#include <hip/hip_runtime.h>

// PillarFeatureNetScatter for MI455X (gfx1250, wave32).
// Reference shapes (fixed): x (B=4, N=100000, C=64) f32,
// indices (B, N) int64 in [0, P), P = 512*512 = 262144.
// Output: (B, C, 512, 512) f32  ==  out[b][c][p] channel-major.

#define P_BINS   (512 * 512)   // 262144
#define P_BYTES  (P_BINS * 4)  // 1048576 -> fits signed 24-bit ioffset
#define N_PTS    100000
#define C_CH     64
#define WAVES_PER_BLOCK 8      // 256 threads = 8 wave32

// ---------------------------------------------------------------------------
// Scatter-add (defined FIRST so the disasm snippet shows it).
// One wave32 per point:
//   lane l loads x[point][2l .. 2l+1] as float2  -> 256B coalesced per point
//   bin index is lane-uniform -> readfirstlane -> SGPR
//   lane l atomically adds into out[b][2l][p] and out[b][2l+1][p]
//   (64 distinct cachelines per point: zero intra-wave atomic conflicts)
// The two atomics are emitted as explicit global_atomic_add_f32 at device
// scope; the second channel is +1 MB which fits the 24-bit ioffset field.
// ---------------------------------------------------------------------------
__global__ __launch_bounds__(256) void pillar_scatter_kernel(
        const float*     __restrict__ x,      // (B*N, C) flat
        const long long* __restrict__ idxs,   // (B*N)    flat int64
        float*           __restrict__ out,    // (B, C, P) flat
        int total_pts) {
    const int lane = (int)(threadIdx.x & 31u);
    const int wave = (int)(threadIdx.x >> 5u);
    const int g    = (int)blockIdx.x * WAVES_PER_BLOCK + wave;   // global point id
    if (g >= total_pts) return;

    const int b = g / N_PTS;                  // batch of this point

    // Lane-uniform bin index -> force into SGPR.
    const int p = __builtin_amdgcn_readfirstlane((int)idxs[g]);

    // Coalesced 256B read of this point's 64 features (8B per lane).
    const float2* xp = (const float2*)(x + (size_t)g * C_CH);

    // gfx1250 prefetch (global_prefetch_b8): pull the next block's point
    // features toward this WGP while the atomics below drain through L2.
    {
        const int gp = g + WAVES_PER_BLOCK;
        if (gp < total_pts) {
            __builtin_prefetch((const char*)(x + (size_t)gp * C_CH) + lane * 8, 0, 3);
        }
    }

    const float2 v = xp[lane];

    if ((unsigned)p < (unsigned)P_BINS) {
        // out[b][2*lane + 0][p]; second atomic hits out[b][2*lane + 1][p]
        // via the +1 MB instruction offset.
        float* o = out + ((size_t)b * C_CH + (size_t)(2 * lane)) * (size_t)P_BINS
                       + (size_t)p;
        // Hardware f32 atomics, no-return (STOREcnt), device scope for
        // cross-WGP coherence.
        asm volatile(
            "global_atomic_add_f32 %0, %1, off scope:SCOPE_DEV\n\t"
            "global_atomic_add_f32 %0, %2, off offset:" "1048576" " scope:SCOPE_DEV"
            :
            : "v"(o), "v"(v.x), "v"(v.y)
            : "memory");
    }
}

// ---------------------------------------------------------------------------
// Zero the 268 MB output with 16-byte stores (bandwidth-limited).
// ---------------------------------------------------------------------------
__global__ __launch_bounds__(256) void pillar_zero_kernel(float4* __restrict__ out,
                                                          int n4) {
    const int stride = (int)(gridDim.x * blockDim.x);
    for (int i = (int)(blockIdx.x * blockDim.x + threadIdx.x); i < n4; i += stride) {
        out[i] = make_float4(0.f, 0.f, 0.f, 0.f);
    }
}

// ---------------------------------------------------------------------------
// Launch
// ---------------------------------------------------------------------------
extern "C" void kernel_launch(void* const* d_in, const int* in_sizes, int n_in,
                              void* d_out, int out_size, void* d_ws, size_t ws_size,
                              hipStream_t stream) {
    const float*     x    = (const float*)d_in[0];      // (B, N, C) f32
    const long long* idxs = (const long long*)d_in[1];  // (B, N)    int64
    float*           out  = (float*)d_out;              // (B, C, 512, 512) f32

    const int total_pts = in_sizes[1];                  // B * N = 400000
    const int n4        = out_size / 4;                 // out_size = 67,108,864 floats

    // Pass 1: zero-fill output (required every call: atomics accumulate and the
    // harness replays the captured graph without re-poisoning buffers).
    {
        dim3 block(256);
        dim3 grid(4096);
        hipLaunchKernelGGL(pillar_zero_kernel, grid, block, 0, stream,
                           (float4*)out, n4);
    }

    // Pass 2: scatter-add, one wave32 per point, 8 points per 256-thread block.
    {
        dim3 block(256);
        dim3 grid((unsigned)((total_pts + WAVES_PER_BLOCK - 1) / WAVES_PER_BLOCK));
        hipLaunchKernelGGL(pillar_scatter_kernel, grid, block, 0, stream,
                           x, idxs, out, total_pts);
    }
}